// pFLNetLayer_73607149519088
// MI455X (gfx1250) — compile-verified
//
#include <hip/hip_runtime.h>
#include <stdint.h>

typedef float v2f __attribute__((ext_vector_type(2)));
typedef float v8f __attribute__((ext_vector_type(8)));
typedef unsigned int v4u __attribute__((ext_vector_type(4)));
typedef int v4i __attribute__((ext_vector_type(4)));
typedef int v8i __attribute__((ext_vector_type(8)));

#define R_RUNS     512
#define N_SAMP     16384
#define FDIM       4
#define ADMM_ITERS 10
#define CH         2048            // chunk elements per array (8 KB)
#define NCH        (N_SAMP / CH)   // 8 chunks
#define TPB1       256

// ---------------------------------------------------------------------------
// TDM: 1-D tensor load of CH f32 elements from global to LDS offset.
// D# group0: count=1, lds_addr, 57-bit global addr, type=2.
// D# group1: data_size=4B, tensor_dim0=CH, tensor_dim1=1, tile_dim0=CH,
//            tile_dim1/2=0 (unused), stride0=CH.
// ---------------------------------------------------------------------------
__device__ __forceinline__ void tdm_load_1d(uint64_t gaddr, uint32_t lds_off) {
  v4u g0;
  g0[0] = 1u;                                             // count=1
  g0[1] = lds_off;                                        // lds_addr (bytes)
  g0[2] = (uint32_t)(gaddr & 0xFFFFFFFFull);              // global_addr[31:0]
  g0[3] = (uint32_t)((gaddr >> 32) & 0x1FFFFFFull)        // global_addr[56:32]
        | (2u << 30);                                     // type=2 (image)
  v8i g1;
  g1[0] = (int)(2u << 16);                                // data_size=2 (4B)
  g1[1] = (int)(((uint32_t)CH & 0xFFFFu) << 16);          // tensor_dim0[15:0]
  g1[2] = (int)(((uint32_t)CH >> 16) | (1u << 16));       // dim0 hi | tensor_dim1=1
  g1[3] = (int)(((uint32_t)CH & 0xFFFFu) << 16);          // tile_dim0 = CH
  g1[4] = 0;                                              // tile_dim1=0, tile_dim2=0
  g1[5] = (int)CH;                                        // tensor_dim0_stride lo
  g1[6] = 0;
  g1[7] = 0;
  v4i z4 = {0, 0, 0, 0};
#if __clang_major__ >= 23
  v8i z8 = {0, 0, 0, 0, 0, 0, 0, 0};
  __builtin_amdgcn_tensor_load_to_lds(g0, g1, z4, z4, z8, 0);
#else
  __builtin_amdgcn_tensor_load_to_lds(g0, g1, z4, z4, 0);
#endif
}

// ---------------------------------------------------------------------------
// Phase 1: per-r moment reduction. One block per r. 256 threads (8 waves).
// Double-buffered TDM streaming: LDS = 2*(8KB x + 8KB Y) = 32 KB dynamic.
//   ws[r*16 + 0..5] = m1..m6  (m0 = N known)
//   ws[r*16 + 6..9] = s0..s3  (sk = sum x^k * y)
// ---------------------------------------------------------------------------
#define XOFF(b) ((uint32_t)((b) * (CH * 4)))
#define YOFF(b) ((uint32_t)(2 * (CH * 4) + (b) * (CH * 4)))

__global__ __launch_bounds__(TPB1) void pfl_moments_kernel(
    const float* __restrict__ x, const float* __restrict__ y,
    float* __restrict__ ws) {
  extern __shared__ char smem[];
  const int r = blockIdx.x;
  const uint64_t gx = (uint64_t)(uintptr_t)(x + (size_t)r * N_SAMP);
  const uint64_t gy = (uint64_t)(uintptr_t)(y + (size_t)r * N_SAMP);

  float acc[10];
#pragma unroll
  for (int j = 0; j < 10; ++j) acc[j] = 0.0f;

  // prologue: wave 0 issues chunks 0 and 1 (4 tensor ops in flight)
  if (threadIdx.x < 32) {
    tdm_load_1d(gx + 0ull * CH * 4, XOFF(0));
    tdm_load_1d(gy + 0ull * CH * 4, YOFF(0));
    tdm_load_1d(gx + 1ull * CH * 4, XOFF(1));
    tdm_load_1d(gy + 1ull * CH * 4, YOFF(1));
  }

  for (int c = 0; c < NCH; ++c) {
    if (threadIdx.x < 32) {
      if (c + 1 < NCH) __builtin_amdgcn_s_wait_tensorcnt(2);  // chunk c done
      else             __builtin_amdgcn_s_wait_tensorcnt(0);
    }
    __syncthreads();  // buffer (c&1) now valid for all waves

    const int b = c & 1;
    const float4* xb = (const float4*)(smem + XOFF(b));
    const float4* yb = (const float4*)(smem + YOFF(b));
#pragma unroll
    for (int i = 0; i < CH / 4 / TPB1; ++i) {
      float4 xv = xb[threadIdx.x + TPB1 * i];
      float4 yv = yb[threadIdx.x + TPB1 * i];
      const float xs[4] = {xv.x, xv.y, xv.z, xv.w};
      const float ys[4] = {yv.x, yv.y, yv.z, yv.w};
#pragma unroll
      for (int k = 0; k < 4; ++k) {
        float x1 = xs[k];
        float x2 = x1 * x1;
        float x3 = x2 * x1;
        float yk = ys[k];
        acc[0] += x1;
        acc[1] += x2;
        acc[2] += x3;
        acc[3] = fmaf(x2, x2, acc[3]);
        acc[4] = fmaf(x2, x3, acc[4]);
        acc[5] = fmaf(x3, x3, acc[5]);
        acc[6] += yk;
        acc[7] = fmaf(x1, yk, acc[7]);
        acc[8] = fmaf(x2, yk, acc[8]);
        acc[9] = fmaf(x3, yk, acc[9]);
      }
    }
    __syncthreads();  // everyone done reading buffer b before overwrite

    if (threadIdx.x < 32 && c + 2 < NCH) {
      tdm_load_1d(gx + (uint64_t)(c + 2) * CH * 4, XOFF(b));
      tdm_load_1d(gy + (uint64_t)(c + 2) * CH * 4, YOFF(b));
    }
  }

  // wave-level shuffle reduce, then cross-wave via LDS (buffers are free now)
#pragma unroll
  for (int j = 0; j < 10; ++j) {
#pragma unroll
    for (int off = 16; off >= 1; off >>= 1)
      acc[j] += __shfl_xor(acc[j], off, 32);
  }
  float* red = (float*)smem;
  const int lane = threadIdx.x & 31;
  const int wid  = threadIdx.x >> 5;
  if (lane == 0) {
#pragma unroll
    for (int j = 0; j < 10; ++j) red[wid * 10 + j] = acc[j];
  }
  __syncthreads();
  if (threadIdx.x < 10) {
    float s = 0.0f;
#pragma unroll
    for (int w = 0; w < TPB1 / 32; ++w) s += red[w * 10 + threadIdx.x];
    ws[r * 16 + threadIdx.x] = s;
  }
}

// ---------------------------------------------------------------------------
// 4x4 inverse (adjugate). tt = XtX + rho*I is SPD -> pinv == inv.
// ---------------------------------------------------------------------------
__device__ __forceinline__ void inv4(const float m[16], float o[16]) {
  float inv[16];
  inv[0]  =  m[5]*m[10]*m[15]-m[5]*m[11]*m[14]-m[9]*m[6]*m[15]+m[9]*m[7]*m[14]+m[13]*m[6]*m[11]-m[13]*m[7]*m[10];
  inv[4]  = -m[4]*m[10]*m[15]+m[4]*m[11]*m[14]+m[8]*m[6]*m[15]-m[8]*m[7]*m[14]-m[12]*m[6]*m[11]+m[12]*m[7]*m[10];
  inv[8]  =  m[4]*m[9]*m[15]-m[4]*m[11]*m[13]-m[8]*m[5]*m[15]+m[8]*m[7]*m[13]+m[12]*m[5]*m[11]-m[12]*m[7]*m[9];
  inv[12] = -m[4]*m[9]*m[14]+m[4]*m[10]*m[13]+m[8]*m[5]*m[14]-m[8]*m[6]*m[13]-m[12]*m[5]*m[10]+m[12]*m[6]*m[9];
  inv[1]  = -m[1]*m[10]*m[15]+m[1]*m[11]*m[14]+m[9]*m[2]*m[15]-m[9]*m[3]*m[14]-m[13]*m[2]*m[11]+m[13]*m[3]*m[10];
  inv[5]  =  m[0]*m[10]*m[15]-m[0]*m[11]*m[14]-m[8]*m[2]*m[15]+m[8]*m[3]*m[14]+m[12]*m[2]*m[11]-m[12]*m[3]*m[10];
  inv[9]  = -m[0]*m[9]*m[15]+m[0]*m[11]*m[13]+m[8]*m[1]*m[15]-m[8]*m[3]*m[13]-m[12]*m[1]*m[11]+m[12]*m[3]*m[9];
  inv[13] =  m[0]*m[9]*m[14]-m[0]*m[10]*m[13]-m[8]*m[1]*m[14]+m[8]*m[2]*m[13]+m[12]*m[1]*m[10]-m[12]*m[2]*m[9];
  inv[2]  =  m[1]*m[6]*m[15]-m[1]*m[7]*m[14]-m[5]*m[2]*m[15]+m[5]*m[3]*m[14]+m[13]*m[2]*m[7]-m[13]*m[3]*m[6];
  inv[6]  = -m[0]*m[6]*m[15]+m[0]*m[7]*m[14]+m[4]*m[2]*m[15]-m[4]*m[3]*m[14]-m[12]*m[2]*m[7]+m[12]*m[3]*m[6];
  inv[10] =  m[0]*m[5]*m[15]-m[0]*m[7]*m[13]-m[4]*m[1]*m[15]+m[4]*m[3]*m[13]+m[12]*m[1]*m[7]-m[12]*m[3]*m[5];
  inv[14] = -m[0]*m[5]*m[14]+m[0]*m[6]*m[13]+m[4]*m[1]*m[14]-m[4]*m[2]*m[13]-m[12]*m[1]*m[6]+m[12]*m[2]*m[5];
  inv[3]  = -m[1]*m[6]*m[11]+m[1]*m[7]*m[10]+m[5]*m[2]*m[11]-m[5]*m[3]*m[10]-m[9]*m[2]*m[7]+m[9]*m[3]*m[6];
  inv[7]  =  m[0]*m[6]*m[11]-m[0]*m[7]*m[10]-m[4]*m[2]*m[11]+m[4]*m[3]*m[10]+m[8]*m[2]*m[7]-m[8]*m[3]*m[6];
  inv[11] = -m[0]*m[5]*m[11]+m[0]*m[7]*m[9]+m[4]*m[1]*m[11]-m[4]*m[3]*m[9]-m[8]*m[1]*m[7]+m[8]*m[3]*m[5];
  inv[15] =  m[0]*m[5]*m[10]-m[0]*m[6]*m[9]-m[4]*m[1]*m[10]+m[4]*m[2]*m[9]+m[8]*m[1]*m[6]-m[8]*m[2]*m[5];
  float det = m[0]*inv[0] + m[1]*inv[4] + m[2]*inv[8] + m[3]*inv[12];
  float rd = 1.0f / det;
#pragma unroll
  for (int i = 0; i < 16; ++i) o[i] = inv[i] * rd;
}

__device__ __forceinline__ void mv4(const float M[16], const float v[4], float o[4]) {
#pragma unroll
  for (int i = 0; i < 4; ++i)
    o[i] = M[i*4+0]*v[0] + M[i*4+1]*v[1] + M[i*4+2]*v[2] + M[i*4+3]*v[3];
}

// ---------------------------------------------------------------------------
// Block-wide sum of a 4-vector over 512 threads (16 waves).
// Wave step: shfl_xor tree. Cross-wave step: the 16 wave-partials are packed
// as the 4x16 B operand (column n = (group g=n/4, feature f=n%4), K slot =
// wave-in-group) and contracted against an all-ones A with
// v_wmma_f32_16x16x4_f32 -- the K-sum IS the reduction; the result is
// invariant to the exact K<->(lane-half,vgpr) mapping since A is all ones.
// Two shfl_xor steps then fold the 4 groups.
// ---------------------------------------------------------------------------
__device__ __forceinline__ void block_sum4(const float in[4], float out[4],
                                           float* P) {
  float a[4] = {in[0], in[1], in[2], in[3]};
#pragma unroll
  for (int j = 0; j < 4; ++j) {
#pragma unroll
    for (int off = 16; off >= 1; off >>= 1)
      a[j] += __shfl_xor(a[j], off, 32);
  }
  const int lane = threadIdx.x & 31;
  const int wid  = threadIdx.x >> 5;
  __syncthreads();                       // previous users of P are done
  if (lane == 0) {
#pragma unroll
    for (int j = 0; j < 4; ++j) P[wid * 4 + j] = a[j];
  }
  __syncthreads();

  const int n   = lane & 15;
  const int g   = n >> 2;
  const int f   = n & 3;
  const int klo = (lane < 16) ? 0 : 2;
  v2f B;
  B[0] = P[(4 * g + klo + 0) * 4 + f];
  B[1] = P[(4 * g + klo + 1) * 4 + f];
  v2f A = {1.0f, 1.0f};
  v8f C = {};
  v8f D = __builtin_amdgcn_wmma_f32_16x16x4_f32(
      /*neg_a=*/false, A, /*neg_b=*/false, B,
      /*c_mod=*/(short)0, C, /*reuse_a=*/false, /*reuse_b=*/false);
  float t = D[0];                        // column (lane&15): sum of 4 waves
  t += __shfl_xor(t, 4, 32);             // fold groups
  t += __shfl_xor(t, 8, 32);
#pragma unroll
  for (int j = 0; j < 4; ++j) out[j] = __shfl(t, j, 32);
}

// ---------------------------------------------------------------------------
// Phase 2: ADMM. One block, 512 threads, thread r owns regression r.
// ---------------------------------------------------------------------------
__global__ __launch_bounds__(512) void pfl_admm_kernel(
    const float* __restrict__ ws, const float* __restrict__ rho_,
    const float* __restrict__ eta_, const float* __restrict__ gamma_,
    const float* __restrict__ theta_, const float* __restrict__ p_,
    const float* __restrict__ tri_l, const float* __restrict__ w0,
    const float* __restrict__ v0, float* __restrict__ out) {
  __shared__ float P[64];
  const int r = threadIdx.x;

  float m[7];
  m[0] = (float)N_SAMP;
#pragma unroll
  for (int j = 1; j < 7; ++j) m[j] = ws[r * 16 + (j - 1)];
  float sy[4];
#pragma unroll
  for (int j = 0; j < 4; ++j) sy[j] = ws[r * 16 + 6 + j];

  const float rho   = rho_[r];
  const float eta   = eta_[r];
  const float theta = theta_[r];
  const float c     = p_[r] * gamma_[r];

  float in4[4] = {c, 0.0f, 0.0f, 0.0f}, red4[4];
  block_sum4(in4, red4, P);
  const float csum = red4[0];

  // Minv = inv(XtX + rho I), constant over iterations
  float A[16], Minv[16];
#pragma unroll
  for (int i = 0; i < 4; ++i)
#pragma unroll
    for (int j = 0; j < 4; ++j) A[i * 4 + j] = m[i + j];
#pragma unroll
  for (int i = 0; i < 4; ++i) A[i * 5] += rho;
  inv4(A, Minv);

  // Binv = inv(tri_l[r] + eta I)
  float Bm[16], Binv[16];
#pragma unroll
  for (int k = 0; k < 16; ++k) Bm[k] = tri_l[r * 16 + k];
#pragma unroll
  for (int i = 0; i < 4; ++i) Bm[i * 5] += eta;
  inv4(Bm, Binv);

  float w[4], v[4], z[4], al[4];
#pragma unroll
  for (int f = 0; f < 4; ++f) {
    w[f]  = w0[f];
    v[f]  = v0[r * 4 + f];
    z[f]  = v[f] - w[f];
    al[f] = 0.0f;
  }

  for (int it = 0; it < ADMM_ITERS; ++it) {
    float tmp[4], zin[4], zo[4], contrib[4];
#pragma unroll
    for (int f = 0; f < 4; ++f) al[f] += theta * (z[f] - v[f] + w[f]);
#pragma unroll
    for (int f = 0; f < 4; ++f) tmp[f] = rho * (w[f] + z[f] + al[f]) + sy[f];
    mv4(Minv, tmp, v);
#pragma unroll
    for (int f = 0; f < 4; ++f) zin[f] = v[f] - w[f] - al[f];
    mv4(Binv, zin, zo);
#pragma unroll
    for (int f = 0; f < 4; ++f) z[f] = eta * zo[f];
#pragma unroll
    for (int f = 0; f < 4; ++f) contrib[f] = c * (v[f] - z[f] - al[f]);
    block_sum4(contrib, red4, P);
#pragma unroll
    for (int f = 0; f < 4; ++f) w[f] = red4[f] / csum;
  }

#pragma unroll
  for (int f = 0; f < 4; ++f) out[r * 4 + f] = v[f];
}

// ---------------------------------------------------------------------------
extern "C" void kernel_launch(void* const* d_in, const int* in_sizes, int n_in,
                              void* d_out, int out_size, void* d_ws,
                              size_t ws_size, hipStream_t stream) {
  const float* x     = (const float*)d_in[0];
  const float* Y     = (const float*)d_in[1];
  const float* rho   = (const float*)d_in[2];
  const float* eta   = (const float*)d_in[3];
  const float* gamma = (const float*)d_in[4];
  const float* theta = (const float*)d_in[5];
  const float* p     = (const float*)d_in[6];
  const float* tri_l = (const float*)d_in[7];
  const float* w0    = (const float*)d_in[8];
  const float* v0    = (const float*)d_in[9];
  float* ws  = (float*)d_ws;
  float* out = (float*)d_out;

  // Phase 1: 512 blocks (one per r), 32 KB dynamic LDS for TDM double buffer.
  pfl_moments_kernel<<<R_RUNS, TPB1, 4 * (CH * 4), stream>>>(x, Y, ws);
  // Phase 2: single 512-thread block.
  pfl_admm_kernel<<<1, 512, 0, stream>>>(ws, rho, eta, gamma, theta, p, tri_l,
                                         w0, v0, out);
}